// weighted_GCN_33320356282899
// MI455X (gfx1250) — compile-verified
//
#include <hip/hip_runtime.h>
#include <hip/hip_bf16.h>
#include <math.h>

// Problem constants (match reference)
#define NN     20000
#define EE     320000
#define TT     4
#define F_IN   128
#define HID    256
#define F_OUT  128
#define ROWS   (NN * TT)          // 80000 flattened (N,T) rows
#define BN_EPS 1e-5f

typedef __attribute__((ext_vector_type(2))) float v2f;
typedef __attribute__((ext_vector_type(8))) float v8f;

// ---------------------------------------------------------------------------
// Edge scatter: agg[dst, t, :] += ew[e, t] * x[src, t, :]
// One float4 feature group per work-item; f32 atomics resolve in L2.
// ---------------------------------------------------------------------------
template <int F>
__global__ void scatter_kernel(const float* __restrict__ x,
                               const int* __restrict__ src,
                               const int* __restrict__ dst,
                               const float* __restrict__ ew,
                               float* __restrict__ agg) {
    constexpr int G = F / 4;  // float4 groups per (edge, t)
    const long long total  = (long long)EE * TT * G;
    const long long stride = (long long)gridDim.x * blockDim.x;
    for (long long i = (long long)blockIdx.x * blockDim.x + threadIdx.x;
         i < total; i += stride) {
        const int g  = (int)(i % G);
        const long long et = i / G;
        const int t  = (int)(et % TT);
        const int e  = (int)(et / TT);
        const int s  = src[e];
        const int d  = dst[e];
        const float w = ew[(long long)e * TT + t];
        const float4 xv = *(const float4*)(x + ((long long)s * TT + t) * F + 4 * g);
        float* o = agg + ((long long)d * TT + t) * F + 4 * g;
        atomicAdd(o + 0, w * xv.x);
        atomicAdd(o + 1, w * xv.y);
        atomicAdd(o + 2, w * xv.z);
        atomicAdd(o + 3, w * xv.w);
    }
}

// ---------------------------------------------------------------------------
// H[m, n] = sum_k A[m, k] * W[k, n] + bias[n]
// f32 WMMA path: V_WMMA_F32_16X16X4_F32, one 16x16 tile per wave32.
// A VGPR layout (16x4 f32):  lane = M%16 (+16 selects K-half), vgpr v -> K = kbase + v + 2*(lane>=16)
// B VGPR layout (4x16 f32):  lane = N%16 (+16 selects K-half), vgpr v -> K = kbase + v + 2*(lane>=16)
// C/D layout (16x16 f32):    vgpr v -> M = v + 8*(lane>=16), N = lane%16
// ---------------------------------------------------------------------------
template <int K, int CN>
__global__ void gemm_bias_wmma(const float* __restrict__ A,
                               const float* __restrict__ W,
                               const float* __restrict__ bias,
                               float* __restrict__ H) {
    const int lane    = threadIdx.x;          // 0..31
    const int half    = lane >> 4;            // 0 or 1
    const int lm      = lane & 15;            // 0..15
    const int rowbase = blockIdx.x * 16;      // M tile
    const int ntile   = blockIdx.y * 8 + threadIdx.y;  // 8 waves/block, one N tile each
    const int ncol    = ntile * 16 + lm;

    const float* arow = A + (size_t)(rowbase + lm) * K + 2 * half;

    v8f acc = {};
    for (int k = 0; k < K; k += 4) {
        // A fragment: 2 consecutive f32 (8B aligned)
        const float2 av = *(const float2*)(arow + k);
        v2f a; a.x = av.x; a.y = av.y;
        // B fragment: W is K x CN row-major
        const float* wp = W + (size_t)(k + 2 * half) * CN + ncol;
        v2f b; b.x = wp[0]; b.y = wp[CN];
        // prefetch next A chunk (speculative, dropped if OOB)
        __builtin_prefetch(arow + k + 16, 0, 1);
        acc = __builtin_amdgcn_wmma_f32_16x16x4_f32(
            /*neg_a=*/false, a, /*neg_b=*/false, b,
            /*c_mod=*/(short)0, acc, /*reuse_a=*/false, /*reuse_b=*/false);
    }

    const float bv = bias[ncol];
#pragma unroll
    for (int v = 0; v < 8; ++v) {
        const int m = v + 8 * half;
        H[(size_t)(rowbase + m) * CN + ncol] = acc[v] + bv;
    }
}

// ---------------------------------------------------------------------------
// Per-channel sum / sum-of-squares over ROWS rows (blockDim.x == C)
// ---------------------------------------------------------------------------
template <int C>
__global__ void stats_kernel(const float* __restrict__ h,
                             float* __restrict__ sum,
                             float* __restrict__ sq) {
    const int c = threadIdx.x;
    float s = 0.0f, q = 0.0f;
    for (int r = blockIdx.x; r < ROWS; r += gridDim.x) {
        const float v = h[(long long)r * C + c];
        s += v;
        q += v * v;
    }
    atomicAdd(&sum[c], s);
    atomicAdd(&sq[c], q);
}

// ---------------------------------------------------------------------------
// In-place BatchNorm (training batch stats, population var) + ReLU
// ---------------------------------------------------------------------------
template <int C>
__global__ void norm_relu_kernel(float* __restrict__ h,
                                 const float* __restrict__ sum,
                                 const float* __restrict__ sq,
                                 const float* __restrict__ gamma,
                                 const float* __restrict__ beta) {
    const float inv_cnt = 1.0f / (float)ROWS;
    const long long total  = (long long)ROWS * C;
    const long long stride = (long long)gridDim.x * blockDim.x;
    for (long long i = (long long)blockIdx.x * blockDim.x + threadIdx.x;
         i < total; i += stride) {
        const int c = (int)(i % C);
        const float mean = sum[c] * inv_cnt;
        const float var  = sq[c] * inv_cnt - mean * mean;
        const float inv  = rsqrtf(var + BN_EPS);
        float v = (h[i] - mean) * inv * gamma[c] + beta[c];
        h[i] = v > 0.0f ? v : 0.0f;
    }
}

// ---------------------------------------------------------------------------
// Launch
// ---------------------------------------------------------------------------
extern "C" void kernel_launch(void* const* d_in, const int* in_sizes, int n_in,
                              void* d_out, int out_size, void* d_ws, size_t ws_size,
                              hipStream_t stream) {
    (void)in_sizes; (void)n_in; (void)out_size; (void)ws_size;

    const float* x   = (const float*)d_in[0];   // (N, T, 128)
    const int*   ei  = (const int*)d_in[1];     // (2, E)
    const float* ew  = (const float*)d_in[2];   // (E, T)
    const float* W0  = (const float*)d_in[3];   // (128, 256)
    const float* b0  = (const float*)d_in[4];
    const float* g0  = (const float*)d_in[5];
    const float* be0 = (const float*)d_in[6];
    const float* W1  = (const float*)d_in[7];   // (256, 128)
    const float* b1  = (const float*)d_in[8];
    const float* g1  = (const float*)d_in[9];
    const float* be1 = (const float*)d_in[10];
    float* out = (float*)d_out;                 // (N, T, 128)

    const int* src = ei;
    const int* dst = ei + EE;

    // Workspace layout
    const size_t AGG_BYTES = (size_t)ROWS * HID * sizeof(float);  // 81.92 MB (max)
    char* ws = (char*)d_ws;
    float* agg   = (float*)ws;                       // scatter accumulator (reused)
    float* h1    = (float*)(ws + AGG_BYTES);         // layer-1 activations (ROWS x 256)
    float* stats = (float*)(ws + 2 * AGG_BYTES);     // 768 floats
    float* sum1 = stats;        // 256
    float* sq1  = stats + 256;  // 256
    float* sum2 = stats + 512;  // 128
    float* sq2  = stats + 640;  // 128

    const dim3 waveBlk(32, 8);  // 8 wave32s per block, one 16x16 WMMA tile per wave

    // ---------------- Layer 1 ----------------
    hipMemsetAsync(agg, 0, (size_t)ROWS * F_IN * sizeof(float), stream);
    scatter_kernel<F_IN><<<4096, 256, 0, stream>>>(x, src, dst, ew, agg);
    gemm_bias_wmma<F_IN, HID><<<dim3(ROWS / 16, HID / 128), waveBlk, 0, stream>>>(agg, W0, b0, h1);
    hipMemsetAsync(sum1, 0, 2 * HID * sizeof(float), stream);
    stats_kernel<HID><<<512, HID, 0, stream>>>(h1, sum1, sq1);
    norm_relu_kernel<HID><<<4096, 256, 0, stream>>>(h1, sum1, sq1, g0, be0);

    // ---------------- Layer 2 ----------------
    hipMemsetAsync(agg, 0, (size_t)ROWS * HID * sizeof(float), stream);
    scatter_kernel<HID><<<4096, 256, 0, stream>>>(h1, src, dst, ew, agg);
    gemm_bias_wmma<HID, F_OUT><<<dim3(ROWS / 16, 1), waveBlk, 0, stream>>>(agg, W1, b1, out);
    hipMemsetAsync(sum2, 0, 2 * F_OUT * sizeof(float), stream);
    stats_kernel<F_OUT><<<512, F_OUT, 0, stream>>>(out, sum2, sq2);
    norm_relu_kernel<F_OUT><<<4096, 256, 0, stream>>>(out, sum2, sq2, g1, be1);
}